// Net_85598698209491
// MI455X (gfx1250) — compile-verified
//
#include <hip/hip_runtime.h>
#include <math.h>

typedef float v2f __attribute__((ext_vector_type(2)));
typedef float v8f __attribute__((ext_vector_type(8)));

#define F_IN 128
#define HID  16
#define NCLS 40

// ---------------- degree / normalization ----------------
__global__ void k_deg_init(float* __restrict__ deg, int n) {
    int i = blockIdx.x * blockDim.x + threadIdx.x;
    if (i < n) deg[i] = 1.0f;                       // self loop
}

__global__ void k_deg_edges(const int* __restrict__ dst, float* __restrict__ deg, int e) {
    int i = blockIdx.x * blockDim.x + threadIdx.x;
    if (i < e) atomicAdd(&deg[dst[i]], 1.0f);
}

__global__ void k_dinv(const float* __restrict__ deg, float* __restrict__ dinv, int n) {
    int i = blockIdx.x * blockDim.x + threadIdx.x;
    if (i < n) dinv[i] = rsqrtf(deg[i]);            // deg >= 1 always
}

// ---------------- layer-1 GEMM: hs1 = (x @ W1) * dinv[row]; acc1 = hs1 ----------------
// One wave computes a 16x16 output tile (HID == 16 -> one col tile), K = 128 in steps of 4.
__global__ void k_gemm1(const float* __restrict__ x, const float* __restrict__ W1,
                        const float* __restrict__ dinv,
                        float* __restrict__ hs1, float* __restrict__ acc1, int n) {
    const int lane = threadIdx.x & 31;
    const int wave = blockIdx.x * (blockDim.x >> 5) + (threadIdx.x >> 5);
    const int row0 = wave * 16;
    if (row0 + 15 >= n) {
        if (row0 >= n) return;                      // whole wave uniform (row0 is wave-uniform)
    }
    const int half = lane >> 4;                     // 0: K={0,1}, 1: K={2,3}
    const int l16  = lane & 15;
    const int arow = row0 + l16;                    // A-matrix row for this lane
    const float* xrow = x + (size_t)arow * F_IN + half * 2;

    v8f acc = {};
#pragma unroll
    for (int k = 0; k < F_IN; k += 4) {
        v2f a;                                      // A 16x4: lane<16 -> K=k,k+1 ; lane>=16 -> K=k+2,k+3
        a.x = xrow[k];
        a.y = xrow[k + 1];
        v2f b;                                      // B 4x16: v0 -> K=k(+2), v1 -> K=k+1(+3), N = l16
        const float* wp = W1 + (size_t)(k + half * 2) * HID + l16;
        b.x = wp[0];
        b.y = wp[HID];
        acc = __builtin_amdgcn_wmma_f32_16x16x4_f32(false, a, false, b,
                                                    (short)0, acc, false, false);
    }
    // D layout: VGPR v -> row = row0 + v + half*8, col = l16
#pragma unroll
    for (int v = 0; v < 8; ++v) {
        const int r = row0 + v + half * 8;
        const float val = acc[v] * dinv[r];
        hs1[(size_t)r * HID + l16]  = val;
        acc1[(size_t)r * HID + l16] = val;          // self-loop term
    }
}

// ---------------- layer-1 scatter: acc1[dst] += hs1[src]  (4 threads/edge, float4 each) ----------------
__global__ void k_scatter1(const int* __restrict__ src, const int* __restrict__ dst,
                           const float* __restrict__ hs1, float* __restrict__ acc1, int e) {
    const long long tid = (long long)blockIdx.x * blockDim.x + threadIdx.x;
    const int eidx = (int)(tid >> 2);
    const int c    = (int)(tid & 3);
    if (eidx >= e) return;
    const int s = src[eidx];
    const int d = dst[eidx];
    const float4 v = *(const float4*)(hs1 + (size_t)s * HID + c * 4);
    float* p = acc1 + (size_t)d * HID + c * 4;
    atomicAdd(p + 0, v.x);
    atomicAdd(p + 1, v.y);
    atomicAdd(p + 2, v.z);
    atomicAdd(p + 3, v.w);
}

// ---------------- h1 = relu(dinv[row]*acc1 + b1) ----------------
__global__ void k_relu1(const float* __restrict__ acc1, const float* __restrict__ dinv,
                        const float* __restrict__ b1, float* __restrict__ h1, int n) {
    const int tid = blockIdx.x * blockDim.x + threadIdx.x;
    const int row = tid >> 4;
    const int f   = tid & 15;
    if (row >= n) return;
    const float v = dinv[row] * acc1[tid] + b1[f];
    h1[tid] = fmaxf(v, 0.0f);
}

// ---------------- layer-2 GEMM: hs2 = (h1 @ W2) * dinv[row]; acc2 = hs2 ----------------
// 3 column tiles of 16 cover C = 40 (last tile partially masked after the WMMAs).
__global__ void k_gemm2(const float* __restrict__ h1, const float* __restrict__ W2,
                        const float* __restrict__ dinv,
                        float* __restrict__ hs2, float* __restrict__ acc2, int n) {
    const int lane = threadIdx.x & 31;
    const int wave = blockIdx.x * (blockDim.x >> 5) + (threadIdx.x >> 5);
    const int rt = wave / 3;
    const int ct = wave - rt * 3;
    const int row0 = rt * 16;
    if (row0 >= n) return;                          // wave-uniform
    const int cb   = ct * 16;
    const int half = lane >> 4;
    const int l16  = lane & 15;
    const int arow = row0 + l16;
    const int col  = cb + l16;
    const bool colok = (col < NCLS);
    const float* hrow = h1 + (size_t)arow * HID + half * 2;

    v8f acc = {};
#pragma unroll
    for (int k = 0; k < HID; k += 4) {
        v2f a;
        a.x = hrow[k];
        a.y = hrow[k + 1];
        const int kk = k + half * 2;
        v2f b;
        b.x = colok ? W2[(size_t)kk * NCLS + col]       : 0.0f;
        b.y = colok ? W2[(size_t)(kk + 1) * NCLS + col] : 0.0f;
        acc = __builtin_amdgcn_wmma_f32_16x16x4_f32(false, a, false, b,
                                                    (short)0, acc, false, false);
    }
    if (colok) {
#pragma unroll
        for (int v = 0; v < 8; ++v) {
            const int r = row0 + v + half * 8;
            const float val = acc[v] * dinv[r];
            hs2[(size_t)r * NCLS + col]  = val;
            acc2[(size_t)r * NCLS + col] = val;     // self-loop term
        }
    }
}

// ---------------- layer-2 scatter: acc2[dst] += hs2[src]  (10 threads/edge, float4 each; 40 = 10*4) ----------------
__global__ void k_scatter2(const int* __restrict__ src, const int* __restrict__ dst,
                           const float* __restrict__ hs2, float* __restrict__ acc2, int e) {
    const long long tid = (long long)blockIdx.x * blockDim.x + threadIdx.x;
    const int eidx = (int)(tid / 10);
    const int c    = (int)(tid - (long long)eidx * 10);
    if (eidx >= e) return;
    const int s = src[eidx];
    const int d = dst[eidx];
    const float4 v = *(const float4*)(hs2 + (size_t)s * NCLS + c * 4);
    float* p = acc2 + (size_t)d * NCLS + c * 4;
    atomicAdd(p + 0, v.x);
    atomicAdd(p + 1, v.y);
    atomicAdd(p + 2, v.z);
    atomicAdd(p + 3, v.w);
}

// ---------------- out = log_softmax(dinv[row]*acc2 + b2) ----------------
__global__ void k_logsoftmax(const float* __restrict__ acc2, const float* __restrict__ dinv,
                             const float* __restrict__ b2, float* __restrict__ out, int n) {
    const int row = blockIdx.x * blockDim.x + threadIdx.x;
    if (row >= n) return;
    const float di = dinv[row];
    const float* p = acc2 + (size_t)row * NCLS;
    float v[NCLS];
    float mx = -INFINITY;
#pragma unroll
    for (int c = 0; c < NCLS; ++c) {
        v[c] = di * p[c] + b2[c];
        mx = fmaxf(mx, v[c]);
    }
    float s = 0.0f;
#pragma unroll
    for (int c = 0; c < NCLS; ++c) s += __expf(v[c] - mx);
    const float lse = mx + __logf(s);
    float* o = out + (size_t)row * NCLS;
#pragma unroll
    for (int c = 0; c < NCLS; ++c) o[c] = v[c] - lse;
}

extern "C" void kernel_launch(void* const* d_in, const int* in_sizes, int n_in,
                              void* d_out, int out_size, void* d_ws, size_t ws_size,
                              hipStream_t stream) {
    const float* x  = (const float*)d_in[0];
    const int*   ei = (const int*)d_in[1];
    const float* W1 = (const float*)d_in[2];
    const float* b1 = (const float*)d_in[3];
    const float* W2 = (const float*)d_in[4];
    const float* b2 = (const float*)d_in[5];
    float* out = (float*)d_out;

    const int n = in_sizes[0] / F_IN;
    const int e = in_sizes[1] / 2;
    const int* src = ei;
    const int* dst = ei + e;

    // workspace layout (256B-aligned chunks); total ~ n*4*114 bytes ~ 91 MB
    char* ws = (char*)d_ws;
    size_t off = 0;
    auto alloc = [&](size_t bytes) -> void* {
        void* p = ws + off;
        off += (bytes + 255) & ~(size_t)255;
        return p;
    };
    float* deg  = (float*)alloc((size_t)n * 4);
    float* dinv = (float*)alloc((size_t)n * 4);
    float* hs1  = (float*)alloc((size_t)n * HID * 4);   // reused as h1 after relu
    float* acc1 = (float*)alloc((size_t)n * HID * 4);
    float* hs2  = (float*)alloc((size_t)n * NCLS * 4);
    float* acc2 = (float*)alloc((size_t)n * NCLS * 4);
    float* h1   = hs1;                                  // alias: hs1 dead once scatter1 done

    const int B = 256;

    k_deg_init<<<(n + B - 1) / B, B, 0, stream>>>(deg, n);
    k_deg_edges<<<(e + B - 1) / B, B, 0, stream>>>(dst, deg, e);
    k_dinv<<<(n + B - 1) / B, B, 0, stream>>>(deg, dinv, n);

    // GEMM1: one wave per 16 rows, 8 waves/block
    {
        const int waves = (n + 15) / 16;
        const int blocks = (waves + 7) / 8;
        k_gemm1<<<blocks, B, 0, stream>>>(x, W1, dinv, hs1, acc1, n);
    }

    {
        const long long threads = (long long)e * 4;
        k_scatter1<<<(unsigned)((threads + B - 1) / B), B, 0, stream>>>(src, dst, hs1, acc1, e);
    }

    k_relu1<<<((n * HID) + B - 1) / B, B, 0, stream>>>(acc1, dinv, b1, h1, n);

    // GEMM2: 3 column tiles per 16-row tile
    {
        const int waves = ((n + 15) / 16) * 3;
        const int blocks = (waves + 7) / 8;
        k_gemm2<<<blocks, B, 0, stream>>>(h1, W2, dinv, hs2, acc2, n);
    }

    {
        const long long threads = (long long)e * 10;
        k_scatter2<<<(unsigned)((threads + B - 1) / B), B, 0, stream>>>(src, dst, hs2, acc2, e);
    }

    k_logsoftmax<<<(n + B - 1) / B, B, 0, stream>>>(acc2, dinv, b2, out, n);
}